// TverskyProjectionLayer_24309514895593
// MI455X (gfx1250) — compile-verified
//
#include <hip/hip_runtime.h>
#include <stdint.h>

// ---------------------------------------------------------------------------
// Tversky projection layer, MI455X (gfx1250, wave32, WMMA + async-to-LDS).
//   B=1024, P=128, F=512, D=256.
//   Pipeline:
//     1) pf    = prototypes @ Wt          (WMMA f32 16x16x4, 16x64 tile/wave)
//     2) xf    = x @ Wt                   (WMMA f32 16x16x4, 16x64 tile/wave)
//     3) inter = relu(xf) @ relu(pf)^T    (WMMA f32 16x16x4, relu on load)
//     4) masked diff pass + final ratio   (async global->LDS double-buffered,
//                                          float4 LDS reads, VALU reduce)
// ---------------------------------------------------------------------------

typedef __attribute__((ext_vector_type(2))) float v2f;
typedef __attribute__((ext_vector_type(8))) float v8f;

constexpr int Bsz = 1024;
constexpr int Psz = 128;
constexpr int Fsz = 512;
constexpr int Dsz = 256;
constexpr float THETA = 1e-7f;

// ---------------------------------------------------------------------------
// GEMM: out[M x N] = act(A[M x K]) @ act(W[N x K])^T, act = relu if RELU.
// One wave computes a 16x64 output tile (4 WMMA accumulators) so the A
// fragment is loaded once per k-step and fed to 4 back-to-back WMMAs.
//
// f32 16x16x4 operand layout (ISA 7.12.2):
//   A: lanes 0-15 v0/v1 = A[m][k+0..1]; lanes 16-31 v0/v1 = A[m][k+2..3]
//   B (= W^T): same striping on n rows -> both are one float2 load at
//   [row][k + 2*half].
// ---------------------------------------------------------------------------
template <bool RELU>
__global__ void gemm_nt_wmma(const float* __restrict__ A,
                             const float* __restrict__ W,
                             float* __restrict__ out,
                             int M, int N, int K) {
  const int wavesPerBlock = blockDim.x >> 5;
  const int wave = blockIdx.x * wavesPerBlock + (threadIdx.x >> 5);
  const int tilesN = N >> 6;  // 64-wide N tiles
  const int tm = wave / tilesN;
  const int tn = wave % tilesN;

  const int lane = threadIdx.x & 31;
  const int half = lane >> 4;  // 0: K pair {0,1}; 1: K pair {2,3}
  const int l = lane & 15;     // row within 16-tile (m for A, n for B)

  const float* aRow = A + (size_t)(tm * 16 + l) * K + 2 * half;
  const float* bRow = W + (size_t)(tn * 64 + l) * K + 2 * half;
  const size_t bStride = (size_t)16 * K;  // next 16-row n-subtile

  v8f acc0 = {}, acc1 = {}, acc2 = {}, acc3 = {};
  for (int k = 0; k < K; k += 4) {
    if ((k & 63) == 0 && k + 64 < K) {  // uniform branch: EXEC stays all-1s
      __builtin_prefetch(aRow + k + 64, 0, 3);
      __builtin_prefetch(bRow + k + 64, 0, 3);
      __builtin_prefetch(bRow + bStride + k + 64, 0, 3);
      __builtin_prefetch(bRow + 2 * bStride + k + 64, 0, 3);
      __builtin_prefetch(bRow + 3 * bStride + k + 64, 0, 3);
    }
    v2f a  = {aRow[k], aRow[k + 1]};
    v2f b0 = {bRow[k], bRow[k + 1]};
    v2f b1 = {bRow[bStride + k], bRow[bStride + k + 1]};
    v2f b2 = {bRow[2 * bStride + k], bRow[2 * bStride + k + 1]};
    v2f b3 = {bRow[3 * bStride + k], bRow[3 * bStride + k + 1]};
    if (RELU) {
      a.x  = fmaxf(a.x, 0.f);  a.y  = fmaxf(a.y, 0.f);
      b0.x = fmaxf(b0.x, 0.f); b0.y = fmaxf(b0.y, 0.f);
      b1.x = fmaxf(b1.x, 0.f); b1.y = fmaxf(b1.y, 0.f);
      b2.x = fmaxf(b2.x, 0.f); b2.y = fmaxf(b2.y, 0.f);
      b3.x = fmaxf(b3.x, 0.f); b3.y = fmaxf(b3.y, 0.f);
    }
    acc0 = __builtin_amdgcn_wmma_f32_16x16x4_f32(false, a, false, b0, (short)0, acc0, false, false);
    acc1 = __builtin_amdgcn_wmma_f32_16x16x4_f32(false, a, false, b1, (short)0, acc1, false, false);
    acc2 = __builtin_amdgcn_wmma_f32_16x16x4_f32(false, a, false, b2, (short)0, acc2, false, false);
    acc3 = __builtin_amdgcn_wmma_f32_16x16x4_f32(false, a, false, b3, (short)0, acc3, false, false);
  }

  // C/D layout: VGPR r -> (M = r + 8*half, N = l)
  float* o = out + (size_t)(tm * 16 + half * 8) * N + (tn * 64 + l);
#pragma unroll
  for (int r = 0; r < 8; ++r) {
    o[(size_t)r * N + 0]  = acc0[r];
    o[(size_t)r * N + 16] = acc1[r];
    o[(size_t)r * N + 32] = acc2[r];
    o[(size_t)r * N + 48] = acc3[r];
  }
}

// ---------------------------------------------------------------------------
// Masked signed-difference reduction + final Tversky ratio.
// Block = 256 threads = one 16(b) x 16(p) output tile. F chunks stream
// through double-buffered LDS via GLOBAL_LOAD_ASYNC_TO_LDS_B128 (ASYNCcnt),
// overlapping the next chunk's fill with the current chunk's reduction.
// ---------------------------------------------------------------------------
constexpr int FCHUNK = 64;
constexpr int FPAD = 4;  // +4 floats/row: conflict-free b128 reads, 16B align
constexpr int NCHUNK = Fsz / FCHUNK;  // 8

__device__ __forceinline__ void async_ld16(uint32_t lds_off, const float* g) {
  asm volatile("global_load_async_to_lds_b128 %0, %1, off"
               :: "v"(lds_off), "v"((uint64_t)(uintptr_t)g)
               : "memory");
}

__global__ void tversky_diff_final(const float* __restrict__ xf,
                                   const float* __restrict__ pf,
                                   const float* __restrict__ inter,
                                   const float* __restrict__ alpha_p,
                                   const float* __restrict__ beta_p,
                                   float* __restrict__ out) {
  __shared__ __align__(16) float sx[2][16][FCHUNK + FPAD];
  __shared__ __align__(16) float sp[2][16][FCHUNK + FPAD];

  const int tid = threadIdx.x;  // 256 threads
  const int bl = tid & 15;      // local b row
  const int pl = tid >> 4;      // local p row
  const int b0 = blockIdx.x * 16;
  const int p0 = blockIdx.y * 16;

  // Async-fill mapping: each thread copies one 16B quad.
  const int lrow = tid >> 4;          // 0..15
  const int lcol = (tid & 15) * 4;    // 0,4,...,60

  auto issue_chunk = [&](int ci, int buf) {
    const int f0 = ci * FCHUNK;
    async_ld16((uint32_t)(uintptr_t)&sx[buf][lrow][lcol],
               xf + (size_t)(b0 + lrow) * Fsz + f0 + lcol);
    async_ld16((uint32_t)(uintptr_t)&sp[buf][lrow][lcol],
               pf + (size_t)(p0 + lrow) * Fsz + f0 + lcol);
  };

  float xmp = 0.0f;  // sum over both>0 of max(xf-pf, 0)
  float pmx = 0.0f;  // sum over both>0 of max(pf-xf, 0)

  issue_chunk(0, 0);
  for (int ci = 0; ci < NCHUNK; ++ci) {
    const int buf = ci & 1;
    if (ci + 1 < NCHUNK) {
      issue_chunk(ci + 1, buf ^ 1);
      // 2 newer async ops outstanding; wait until chunk ci's 2 are done.
      asm volatile("s_wait_asynccnt 0x2" ::: "memory");
    } else {
      asm volatile("s_wait_asynccnt 0x0" ::: "memory");
    }
    __syncthreads();  // chunk ci visible to all waves

#pragma unroll 4
    for (int c = 0; c < FCHUNK; c += 4) {
      const float4 xv = *(const float4*)&sx[buf][bl][c];
      const float4 pv = *(const float4*)&sp[buf][pl][c];
      {
        if (xv.x > 0.f && pv.x > 0.f) { float d = xv.x - pv.x; xmp += fmaxf(d, 0.f); pmx += fmaxf(-d, 0.f); }
        if (xv.y > 0.f && pv.y > 0.f) { float d = xv.y - pv.y; xmp += fmaxf(d, 0.f); pmx += fmaxf(-d, 0.f); }
        if (xv.z > 0.f && pv.z > 0.f) { float d = xv.z - pv.z; xmp += fmaxf(d, 0.f); pmx += fmaxf(-d, 0.f); }
        if (xv.w > 0.f && pv.w > 0.f) { float d = xv.w - pv.w; xmp += fmaxf(d, 0.f); pmx += fmaxf(-d, 0.f); }
      }
    }
    __syncthreads();  // all reads of chunk ci done before its buffer is refilled
  }

  const float alpha = alpha_p[0];
  const float beta = beta_p[0];
  const size_t idx = (size_t)(b0 + bl) * Psz + (p0 + pl);
  const float it = inter[idx];
  out[idx] = it / (it + alpha * xmp + beta * pmx + THETA);
}

extern "C" void kernel_launch(void* const* d_in, const int* in_sizes, int n_in,
                              void* d_out, int out_size, void* d_ws, size_t ws_size,
                              hipStream_t stream) {
  const float* x     = (const float*)d_in[0];  // [B, D]
  const float* prot  = (const float*)d_in[1];  // [P, D]
  const float* fb    = (const float*)d_in[2];  // [F, D]
  const float* alpha = (const float*)d_in[3];  // scalar
  const float* beta  = (const float*)d_in[4];  // scalar
  float* out = (float*)d_out;                  // [B, P]

  // Workspace: pf [P*F] | xf [B*F] | inter [B*P]  (~2.9 MB total)
  float* pf    = (float*)d_ws;
  float* xf    = pf + (size_t)Psz * Fsz;
  float* inter = xf + (size_t)Bsz * Fsz;

  // 4 waves per 128-thread block; each wave owns a 16x64 tile.
  {
    const int waves = (Psz / 16) * (Fsz / 64);  // 64
    gemm_nt_wmma<false><<<waves / 4, 128, 0, stream>>>(prot, fb, pf, Psz, Fsz, Dsz);
  }
  {
    const int waves = (Bsz / 16) * (Fsz / 64);  // 512
    gemm_nt_wmma<false><<<waves / 4, 128, 0, stream>>>(x, fb, xf, Bsz, Fsz, Dsz);
  }
  {
    const int waves = (Bsz / 16) * (Psz / 64);  // 128
    gemm_nt_wmma<true><<<waves / 4, 128, 0, stream>>>(xf, pf, inter, Bsz, Psz, Fsz);
  }
  {
    dim3 grid(Bsz / 16, Psz / 16);  // 64 x 8
    tversky_diff_final<<<grid, 256, 0, stream>>>(xf, pf, inter, alpha, beta, out);
  }
}